// ClusterLayer_24524263260901
// MI455X (gfx1250) — compile-verified
//
#include <hip/hip_runtime.h>

typedef float v2f __attribute__((ext_vector_type(2)));
typedef float v8f __attribute__((ext_vector_type(8)));

constexpr int Nn = 512, Ll = 512, Ff = 64, Hh = 128;
constexpr int WST = 68;           // padded LDS row stride (dwords): conflict-free, 16B-aligned rows
constexpr float LN_EPS = 1e-5f;

__global__ __launch_bounds__(256) void pn_init_g(float* __restrict__ g) {
  int i = blockIdx.x * 256 + threadIdx.x;
  if (i < Nn * Hh) g[i] = 0.0f;
}

__global__ __launch_bounds__(256) void pn_main(
    const float* __restrict__ x, const int* __restrict__ mask,
    const float* __restrict__ W, const float* __restrict__ b,
    const float* __restrict__ gamma, const float* __restrict__ beta,
    float* __restrict__ out, float* __restrict__ g) {
  __shared__ float Wl[Hh * WST];           // 34816 B
  __shared__ float bl[Hh], gml[Hh], btl[Hh];
  __shared__ int   ml[128];
  __shared__ int   gmax[Hh];

  const int t = threadIdx.x;
  const int rows0 = blockIdx.x * 128;      // 128 point-rows per block, all same n
  const int n = rows0 >> 9;                // rows0 / L

  // stage W[128][64] row-major into padded LDS: 32 floats per thread
  {
    const int r  = t >> 1;
    const int c0 = (t & 1) * 32;
    const float4* __restrict__ Wv = (const float4*)(W + r * Ff + c0);
    float4* Wd = (float4*)(Wl + r * WST + c0);
    #pragma unroll
    for (int i = 0; i < 8; ++i) Wd[i] = Wv[i];
  }
  if (t < Hh)  { bl[t] = b[t]; gml[t] = gamma[t]; btl[t] = beta[t]; gmax[t] = 0; }
  if (t < 128) { ml[t] = mask[rows0 + t]; }
  __syncthreads();

  const int lane = t & 31;
  const int wv   = t >> 5;          // wave id 0..7 -> 16-row tile each
  const int ln   = lane & 15;
  const int half = lane >> 4;       // half 0: K{0,1}, rows 0-7 of C; half 1: K{2,3}, rows 8-15

  v8f acc[8] = {};                  // 8 column tiles of 16 -> H = 128

  // A fragment source: row (rows0 + wv*16 + ln), K offset 4k + 2*half, float2 along K
  const float* __restrict__ xrow =
      x + (size_t)(rows0 + wv * 16 + ln) * Ff + 2 * half;
  const int kb = 2 * half;

  #pragma unroll 4
  for (int k = 0; k < 16; ++k) {
    v2f a = *(const v2f*)(xrow + 4 * k);
    #pragma unroll
    for (int j = 0; j < 8; ++j) {
      // B[kk][col] = W[col][kk]; col = j*16 + ln, float2 along K from padded LDS
      v2f bb = *(const v2f*)&Wl[(j * 16 + ln) * WST + 4 * k + kb];
      acc[j] = __builtin_amdgcn_wmma_f32_16x16x4_f32(
          false, a, false, bb, (short)0, acc[j], false, false);
    }
  }

  // + bias (per column, same across rows)
  #pragma unroll
  for (int j = 0; j < 8; ++j) acc[j] = acc[j] + bl[j * 16 + ln];

  // LayerNorm stats per row m = v + 8*half: sum over j in-register,
  // then butterfly over the 16 lanes of this half (cols live across lanes).
  float mu[8], rs[8];
  #pragma unroll
  for (int v = 0; v < 8; ++v) {
    float s = 0.f, q = 0.f;
    #pragma unroll
    for (int j = 0; j < 8; ++j) { float h = acc[j][v]; s += h; q += h * h; }
    #pragma unroll
    for (int m = 1; m <= 8; m <<= 1) {
      s += __shfl_xor(s, m, 32);
      q += __shfl_xor(q, m, 32);
    }
    float mean = s * (1.0f / Hh);
    float var  = q * (1.0f / Hh) - mean * mean;
    mu[v] = mean;
    rs[v] = rsqrtf(var + LN_EPS);
  }

  // normalize + ReLU, store y, masked column max
  const size_t obase = (size_t)(rows0 + wv * 16 + 8 * half) * (2 * Hh);
  #pragma unroll
  for (int j = 0; j < 8; ++j) {
    const int col = j * 16 + ln;
    const float ga = gml[col], be = btl[col];
    float mx = 0.f;
    #pragma unroll
    for (int v = 0; v < 8; ++v) {
      float y = (acc[j][v] - mu[v]) * rs[v] * ga + be;
      y = fmaxf(y, 0.f);
      out[obase + (size_t)v * (2 * Hh) + col] = y;
      if (ml[wv * 16 + 8 * half + v]) mx = fmaxf(mx, y);
    }
    mx = fmaxf(mx, __shfl_xor(mx, 16, 32));   // fold row-halves
    if (half == 0) atomicMax(&gmax[col], __float_as_int(mx));
  }
  __syncthreads();
  // one global atomic-max per column per block (values are >= 0 -> int max == float max)
  if (t < Hh) atomicMax((int*)&g[n * Hh + t], gmax[t]);
}

__global__ __launch_bounds__(256) void pn_bcast(const float* __restrict__ g,
                                                float* __restrict__ out) {
  int idx = blockIdx.x * 256 + threadIdx.x;   // N*L*(H/4) threads
  int row = idx >> 5;                         // point row
  int c   = idx & 31;                         // float4 column within H
  const float4 gv = ((const float4*)g)[((row >> 9) << 5) + c];
  ((float4*)out)[(size_t)row * 64 + 32 + c] = gv;  // out[row, H + 4c .. H + 4c+3]
}

extern "C" void kernel_launch(void* const* d_in, const int* in_sizes, int n_in,
                              void* d_out, int out_size, void* d_ws, size_t ws_size,
                              hipStream_t stream) {
  const float* x     = (const float*)d_in[0];
  const int*   mask  = (const int*)d_in[1];
  const float* W     = (const float*)d_in[2];
  const float* b     = (const float*)d_in[3];
  const float* gamma = (const float*)d_in[4];
  const float* beta  = (const float*)d_in[5];

  float* out = (float*)d_out;
  float* g   = out + (size_t)Nn * Ll * 2 * Hh;   // g appended after [N,L,2H]

  pn_init_g<<<(Nn * Hh + 255) / 256, 256, 0, stream>>>(g);
  pn_main<<<(Nn * Ll) / 128, 256, 0, stream>>>(x, mask, W, b, gamma, beta, out, g);
  pn_bcast<<<(Nn * Ll * (Hh / 4)) / 256, 256, 0, stream>>>(g, out);
}